// FixedScatteringNetwork_80582176407640
// MI455X (gfx1250) — compile-verified
//
#include <hip/hip_runtime.h>
#include <hip/hip_bf16.h>

typedef float v2f __attribute__((ext_vector_type(2)));
typedef float v8f __attribute__((ext_vector_type(8)));
typedef unsigned int v4u __attribute__((ext_vector_type(4)));
typedef int v8i __attribute__((ext_vector_type(8)));
typedef int v4i __attribute__((ext_vector_type(4)));

#define NPIX 128
#define NN   16384      // 128*128
#define BATCH 64
#define PI_F 3.14159265358979323846f

// LDS strip layout: 16 rows x 129 dwords (padded; +1 dword/row kills bank conflicts)
#define SROW 129
#define SIM_OFF 2112            // 16*132 floats (real strip needs 16*129); imag strip starts here (byte off 8448)

// -------- workspace layout (floats) --------
static constexpr size_t OFF_FRE  = 0;                      // F real       [128x128]
static constexpr size_t OFF_FIM  = OFF_FRE  + NN;          // F imag
static constexpr size_t OFF_FIRE = OFF_FIM  + NN;          // Finv real (conj(F)/N)
static constexpr size_t OFF_FIIM = OFF_FIRE + NN;          // Finv imag
static constexpr size_t OFF_PSI  = OFF_FIIM + NN;          // PSI [16][128x128]
static constexpr size_t OFF_COEF = OFF_PSI  + 16*(size_t)NN; // coeffs [64][11] (pad 1024)
static constexpr size_t OFF_IHRE = OFF_COEF + 1024;        // fft2(img) re [64]
static constexpr size_t OFF_IHIM = OFF_IHRE + 64*(size_t)NN;
static constexpr size_t OFF_U1   = OFF_IHIM + 64*(size_t)NN;  // u1 [l*64+b] real, 256 planes
static constexpr size_t OFF_UHRE = OFF_U1   + 256*(size_t)NN; // u1_hat re, 256 planes
static constexpr size_t OFF_UHIM = OFF_UHRE + 256*(size_t)NN;
static constexpr size_t OFF_TRE  = OFF_UHIM + 256*(size_t)NN; // ping temp, 256 planes
static constexpr size_t OFF_TIM  = OFF_TRE  + 256*(size_t)NN;

// -------- fp32 WMMA wrapper (V_WMMA_F32_16X16X4_F32) --------
__device__ __forceinline__ v8f wmma_f32(v2f a, v2f b, v8f c) {
  return __builtin_amdgcn_wmma_f32_16x16x4_f32(false, a, false, b, (short)0, c, false, false);
}

// -------- TDM: 1-D copy of 2048 dwords global->LDS with 1-dword pad per 128 --------
#if __has_builtin(__builtin_amdgcn_tensor_load_to_lds)
__device__ __forceinline__ void tdm_strip_load(const float* src, unsigned lds_byte_off) {
  unsigned long long ga = (unsigned long long)(size_t)src;
  v4u g0;
  g0.x = 1u;                                   // count=1, user descriptor
  g0.y = lds_byte_off;                         // D#.lds_addr
  g0.z = (unsigned)(ga & 0xFFFFFFFFu);         // global_addr[31:0]
  g0.w = (unsigned)((ga >> 32) & 0x01FFFFFFu)  // global_addr[56:32]
       | (2u << 30);                           // type=2 ("image")
  v8i g1;
  g1[0] = (int)((2u << 16)                     // data_size = 4B
              | (1u << 20)                     // pad_enable
              | (6u << 22));                   // pad_interval: 128 dwords
  g1[1] = (int)(2048u << 16);                  // tensor_dim0[15:0] = 2048 (bits 63:48)
  g1[2] = (int)(1u << 16);                     // tensor_dim0 hi=0, tensor_dim1 lo=1
  g1[3] = (int)(2048u << 16);                  // tile_dim0 = 2048 (bits 127:112)
  g1[4] = 0;                                   // tile_dim1=0 (1-D), tile_dim2=0
  g1[5] = 2048;                                // tensor_dim0_stride[31:0]
  g1[6] = 0;
  g1[7] = 0;
  v4i z4 = {0, 0, 0, 0};
  v8i z8 = {0, 0, 0, 0, 0, 0, 0, 0};
  // 6-arg form (clang-23 / therock headers): (g0, g1, g2, g3, g4, cpol)
  __builtin_amdgcn_tensor_load_to_lds(g0, g1, z4, z4, z8, 0);
}
#endif

// Stage one 16x128 complex strip (row-major contiguous in global) into padded LDS.
__device__ __forceinline__ void stage_strip(const float* Re, const float* Im,
                                            int mTile, float* smem) {
#if __has_builtin(__builtin_amdgcn_tensor_load_to_lds)
  (void)smem;
  if ((threadIdx.x >> 5) == 0) {               // wave 0 only (uniform branch)
    tdm_strip_load(Re + (size_t)mTile * 2048, 0u);
    tdm_strip_load(Im + (size_t)mTile * 2048, (unsigned)(SIM_OFF * 4));
    __builtin_amdgcn_s_wait_tensorcnt(0);
  }
  __syncthreads();
#else
  for (int i = threadIdx.x; i < 2048; i += 256) {
    int r = i >> 7, c = i & 127;
    smem[r * SROW + c]           = Re[(size_t)mTile * 2048 + i];
    smem[SIM_OFF + r * SROW + c] = Im[(size_t)mTile * 2048 + i];
  }
  __syncthreads();
#endif
}

// -------- constants: DFT matrices + Morlet bank --------
__global__ __launch_bounds__(256) void init_const_kernel(float* __restrict__ ws) {
  int idx = blockIdx.x * 256 + threadIdx.x;
  if (idx >= NN) return;
  int row = idx >> 7, col = idx & 127;
  float theta = (2.0f * PI_F / 128.0f) * (float)((row * col) & 127);
  float c = cosf(theta), s = sinf(theta);
  ws[OFF_FRE  + idx] =  c;
  ws[OFF_FIM  + idx] = -s;
  ws[OFF_FIRE + idx] =  c * (1.0f / 128.0f);
  ws[OFF_FIIM + idx] =  s * (1.0f / 128.0f);
  float fx = (float)((row < 64) ? row : row - 128) * (2.0f * PI_F / 128.0f);
  float fy = (float)((col < 64) ? col : col - 128) * (2.0f * PI_F / 128.0f);
  #pragma unroll
  for (int j = 0; j < 4; ++j) {
    float k0 = 3.0f * PI_F / 4.0f / (float)(1 << j);
    float sg = 0.8f * (float)(1 << j);
    float s2 = 0.5f * sg * sg;
    float beta = expf(-s2 * k0 * k0);
    float gz = expf(-s2 * (fx * fx + fy * fy));
    #pragma unroll
    for (int l = 0; l < 4; ++l) {
      float th = PI_F * (float)l * 0.25f;
      float dx = fx - k0 * cosf(th), dy = fy - k0 * sinf(th);
      ws[OFF_PSI + (size_t)(j * 4 + l) * NN + idx] =
          expf(-s2 * (dx * dx + dy * dy)) - beta * gz;
    }
  }
}

__global__ void zero_kernel(float* __restrict__ p, int n) {
  int i = blockIdx.x * 256 + threadIdx.x;
  if (i < n) p[i] = 0.0f;
}

__global__ __launch_bounds__(256) void s0_kernel(const float* __restrict__ img,
                                                 float* __restrict__ coeffs) {
  int b = blockIdx.x;
  const float* p = img + (size_t)b * NN;
  float sum = 0.0f;
  for (int i = threadIdx.x; i < NN; i += 256) sum += p[i];
  for (int off = 16; off > 0; off >>= 1) sum += __shfl_xor(sum, off, 32);
  __shared__ float sb[8];
  if ((threadIdx.x & 31) == 0) sb[threadIdx.x >> 5] = sum;
  __syncthreads();
  if (threadIdx.x == 0) {
    float t = 0.0f;
    #pragma unroll
    for (int w = 0; w < 8; ++w) t += sb[w];
    coeffs[b * 11 + 0] = t * (1.0f / NN);
  }
}

// -------- left matmul: T = M · X --------
// grid = (8 mTiles, planes), block = 256 (8 waves, one 16x16 tile each).
// A-strip (M rows) staged to LDS via TDM; B (X columns) streamed from global.
template <int XMODE>   // 0: X real; 1: X complex; 2: X complex * real filter W
__global__ __launch_bounds__(256)
void dft_left_kernel(const float* __restrict__ Mre, const float* __restrict__ Mim,
                     const float* __restrict__ Xre, const float* __restrict__ Xim,
                     const float* __restrict__ W,
                     float* __restrict__ Tre, float* __restrict__ Tim) {
  __shared__ float smem[2 * SIM_OFF];
  const int lane = threadIdx.x & 31;
  const int nTile = threadIdx.x >> 5;
  const int mTile = blockIdx.x;
  const int p = blockIdx.y;
  const int half = lane >> 4;
  const int ml = lane & 15;
  const int n = nTile * 16 + ml;

  stage_strip(Mre, Mim, mTile, smem);

  const float* xre = Xre + (size_t)p * NN;
  const float* xim = (XMODE >= 1) ? (Xim + (size_t)p * NN) : nullptr;
  const float* sRe = smem + ml * SROW;
  const float* sIm = smem + SIM_OFF + ml * SROW;

  v8f accRR = {}, accRI = {}, accIR = {}, accII = {};
  for (int k0 = 0; k0 < NPIX; k0 += 4) {
    const int kb = k0 + (half << 1);
    v2f aR, aI, bR;
    aR.x = sRe[kb]; aR.y = sRe[kb + 1];      // ds_load_b64, conflict-free (129 stride)
    aI.x = sIm[kb]; aI.y = sIm[kb + 1];
    float w0 = 1.0f, w1 = 1.0f;
    if (XMODE == 2) { w0 = W[kb * NPIX + n]; w1 = W[(kb + 1) * NPIX + n]; }
    bR.x = xre[kb * NPIX + n] * w0; bR.y = xre[(kb + 1) * NPIX + n] * w1;
    accRR = wmma_f32(aR, bR, accRR);
    accIR = wmma_f32(aI, bR, accIR);
    if (XMODE >= 1) {
      v2f bI;
      bI.x = xim[kb * NPIX + n] * w0; bI.y = xim[(kb + 1) * NPIX + n] * w1;
      accRI = wmma_f32(aR, bI, accRI);
      accII = wmma_f32(aI, bI, accII);
    }
  }
  float* tre = Tre + (size_t)p * NN;
  float* tim = Tim + (size_t)p * NN;
  const int mbase = mTile * 16 + (half << 3);
  #pragma unroll
  for (int r = 0; r < 8; ++r) {
    const int mm = mbase + r;
    tre[mm * NPIX + n] = accRR[r] - accII[r];
    tim[mm * NPIX + n] = accIR[r] + accRI[r];
  }
}

// -------- right matmul: Y = T · M, fused epilogue --------
// A-strip (T rows, block-shared) staged to LDS via TDM; B (M columns) from global.
template <int OUT>  // 0: store complex; 1: store |Y| + atomic mean; 2: atomic mean only
__global__ __launch_bounds__(256)
void dft_right_kernel(const float* __restrict__ Tre, const float* __restrict__ Tim,
                      const float* __restrict__ Mre, const float* __restrict__ Mim,
                      float* __restrict__ Yre, float* __restrict__ Yim,
                      float* __restrict__ U, int uPlaneOff,
                      float* __restrict__ coeffs, int cidx, float scale) {
  __shared__ float smem[2 * SIM_OFF];
  const int lane = threadIdx.x & 31;
  const int nTile = threadIdx.x >> 5;
  const int mTile = blockIdx.x;
  const int p = blockIdx.y;
  const int half = lane >> 4;
  const int ml = lane & 15;
  const int n = nTile * 16 + ml;

  // T-strip of this plane: rows mTile*16..+15, contiguous 8KB each for re/im
  stage_strip(Tre + (size_t)p * NN, Tim + (size_t)p * NN, mTile, smem);

  const float* sRe = smem + ml * SROW;
  const float* sIm = smem + SIM_OFF + ml * SROW;

  v8f accRR = {}, accRI = {}, accIR = {}, accII = {};
  for (int k0 = 0; k0 < NPIX; k0 += 4) {
    const int kb = k0 + (half << 1);
    v2f aR, aI, bR, bI;
    aR.x = sRe[kb]; aR.y = sRe[kb + 1];
    aI.x = sIm[kb]; aI.y = sIm[kb + 1];
    bR.x = Mre[kb * NPIX + n]; bR.y = Mre[(kb + 1) * NPIX + n];
    bI.x = Mim[kb * NPIX + n]; bI.y = Mim[(kb + 1) * NPIX + n];
    accRR = wmma_f32(aR, bR, accRR);
    accII = wmma_f32(aI, bI, accII);
    accRI = wmma_f32(aR, bI, accRI);
    accIR = wmma_f32(aI, bR, accIR);
  }
  const int mbase = mTile * 16 + (half << 3);
  if (OUT == 0) {
    float* yre = Yre + (size_t)p * NN;
    float* yim = Yim + (size_t)p * NN;
    #pragma unroll
    for (int r = 0; r < 8; ++r) {
      const int mm = mbase + r;
      yre[mm * NPIX + n] = accRR[r] - accII[r];
      yim[mm * NPIX + n] = accIR[r] + accRI[r];
    }
  } else {
    float* up = (OUT == 1) ? (U + (size_t)(uPlaneOff + p) * NN) : nullptr;
    float partial = 0.0f;
    #pragma unroll
    for (int r = 0; r < 8; ++r) {
      float re = accRR[r] - accII[r];
      float im = accIR[r] + accRI[r];
      float mag = sqrtf(re * re + im * im);
      if (OUT == 1) up[(mbase + r) * NPIX + n] = mag;
      partial += mag;
    }
    for (int off = 16; off > 0; off >>= 1) partial += __shfl_xor(partial, off, 32);
    if (lane == 0) atomicAdd(&coeffs[(p & 63) * 11 + cidx], partial * scale);
  }
}

// -------- tiny MLP head --------
__global__ void mlp_kernel(const float* __restrict__ coeffs,
                           const float* __restrict__ w1, const float* __restrict__ b1,
                           const float* __restrict__ w2, const float* __restrict__ b2,
                           float* __restrict__ out) {
  int b = threadIdx.x;
  if (b >= BATCH) return;
  float c[11];
  #pragma unroll
  for (int i = 0; i < 11; ++i) c[i] = coeffs[b * 11 + i];
  float h[4];
  #pragma unroll
  for (int i = 0; i < 4; ++i) {
    float a = b1[i];
    #pragma unroll
    for (int j = 0; j < 11; ++j) a += w1[i * 11 + j] * c[j];
    h[i] = fmaxf(a, 0.0f);
  }
  #pragma unroll
  for (int o = 0; o < 10; ++o) {
    float a = b2[o];
    #pragma unroll
    for (int i = 0; i < 4; ++i) a += w2[o * 4 + i] * h[i];
    out[b * 10 + o] = 1.0f / (1.0f + expf(-a));
  }
}

extern "C" void kernel_launch(void* const* d_in, const int* in_sizes, int n_in,
                              void* d_out, int out_size, void* d_ws, size_t ws_size,
                              hipStream_t stream) {
  (void)in_sizes; (void)n_in; (void)out_size; (void)ws_size;
  const float* img  = (const float*)d_in[0];
  const float* fc1w = (const float*)d_in[1];
  const float* fc1b = (const float*)d_in[2];
  const float* fc2w = (const float*)d_in[3];
  const float* fc2b = (const float*)d_in[4];
  float* ws = (float*)d_ws;

  float* Fre  = ws + OFF_FRE;  float* Fim  = ws + OFF_FIM;
  float* FIre = ws + OFF_FIRE; float* FIim = ws + OFF_FIIM;
  float* PSI  = ws + OFF_PSI;
  float* coef = ws + OFF_COEF;
  float* IHre = ws + OFF_IHRE; float* IHim = ws + OFF_IHIM;
  float* U1   = ws + OFF_U1;
  float* UHre = ws + OFF_UHRE; float* UHim = ws + OFF_UHIM;
  float* Tre  = ws + OFF_TRE;  float* Tim  = ws + OFF_TIM;

  dim3 blk(256);

  init_const_kernel<<<64, blk, 0, stream>>>(ws);
  zero_kernel<<<4, blk, 0, stream>>>(coef, 1024);
  s0_kernel<<<BATCH, blk, 0, stream>>>(img, coef);

  // Stage A: i_hat = F · img · F   (64 planes)
  dft_left_kernel<0><<<dim3(8, 64), blk, 0, stream>>>(Fre, Fim, img, nullptr, nullptr, Tre, Tim);
  dft_right_kernel<0><<<dim3(8, 64), blk, 0, stream>>>(Tre, Tim, Fre, Fim, IHre, IHim,
                                                       nullptr, 0, nullptr, 0, 0.0f);

  int pairIdx = 0;
  for (int j1 = 0; j1 < 4; ++j1) {
    // Stage B: u1[l][b] = |Fi·(i_hat ⊙ PSI[j1,l])·Fi|, fused s1 mean
    for (int l = 0; l < 4; ++l) {
      const float* Wf = PSI + (size_t)(j1 * 4 + l) * NN;
      dft_left_kernel<2><<<dim3(8, 64), blk, 0, stream>>>(FIre, FIim, IHre, IHim, Wf, Tre, Tim);
      dft_right_kernel<1><<<dim3(8, 64), blk, 0, stream>>>(Tre, Tim, FIre, FIim,
                                                           nullptr, nullptr, U1, l * 64,
                                                           coef, 1 + j1, 1.0f / (4.0f * NN));
    }
    // Stage C: u1_hat = F · u1 · F   (256 planes)
    dft_left_kernel<0><<<dim3(8, 256), blk, 0, stream>>>(Fre, Fim, U1, nullptr, nullptr, Tre, Tim);
    dft_right_kernel<0><<<dim3(8, 256), blk, 0, stream>>>(Tre, Tim, Fre, Fim, UHre, UHim,
                                                          nullptr, 0, nullptr, 0, 0.0f);
    // Stage D: s2 pair means, fields never materialized
    for (int j2 = j1 + 1; j2 < 4; ++j2, ++pairIdx) {
      for (int l2 = 0; l2 < 4; ++l2) {
        const float* Wf = PSI + (size_t)(j2 * 4 + l2) * NN;
        dft_left_kernel<2><<<dim3(8, 256), blk, 0, stream>>>(FIre, FIim, UHre, UHim, Wf, Tre, Tim);
        dft_right_kernel<2><<<dim3(8, 256), blk, 0, stream>>>(Tre, Tim, FIre, FIim,
                                                              nullptr, nullptr, nullptr, 0,
                                                              coef, 5 + pairIdx,
                                                              1.0f / (16.0f * NN));
      }
    }
  }

  mlp_kernel<<<1, 64, 0, stream>>>(coef, fc1w, fc1b, fc2w, fc2b, (float*)d_out);
}